// TemporalMix_56066503082020
// MI455X (gfx1250) — compile-verified
//
#include <hip/hip_runtime.h>
#include <stdint.h>

typedef __attribute__((ext_vector_type(4))) float v4f;  // true clang vector (NT-store ok)

// ---------------- problem constants ----------------
constexpr int Bn   = 64;
constexpr int Tn   = 16;
constexpr int CHW  = 3 * 112 * 112;        // 37632 floats per (b,t) row
constexpr int ROW_V4 = CHW / 4;            // 9408 float4 per row
constexpr int CHUNKS_PER_ROW = ROW_V4 / 32;// 294 wave-chunks (512B each)
constexpr int NROWS   = Bn * Tn;           // 1024
constexpr int NCHUNKS = NROWS * CHUNKS_PER_ROW; // 301056
constexpr int DEPTH   = 8;                 // async pipeline depth per wave

// ---------------- small hash -> uniform [0,1) ----------------
__device__ __forceinline__ float hash_uniform(uint32_t s, uint32_t t, uint32_t i) {
    uint32_t h = s * 0x9E3779B9u ^ (t * 0x85EBCA6Bu) ^ (i * 0xC2B2AE35u) ^ 0x2545F491u;
    h ^= h >> 16; h *= 0x7FEB352Du;
    h ^= h >> 15; h *= 0x846CA68Bu;
    h ^= h >> 16;
    return (float)(h >> 8) * (1.0f / 16777216.0f);
}

// ---------------- permutation kernel ----------------
// One block per timestep t (64 threads). Rank-sort keys0/keys1, invert order2,
// src[t, order1_pos] mapping collapses to: src[t,i] = inv1[rank0(i)].
__global__ void temporalmix_perm_kernel(const int* __restrict__ y,
                                        int* __restrict__ src,
                                        float* __restrict__ y_out) {
    __shared__ float k0[Bn];
    __shared__ float k1[Bn];
    __shared__ int   inv1[Bn];

    const int t = blockIdx.x;
    const int i = threadIdx.x;

    const float yb = (float)y[i];
    k0[i] = yb + hash_uniform(0u, (uint32_t)t, (uint32_t)i);
    k1[i] = yb + hash_uniform(1u, (uint32_t)t, (uint32_t)i);
    __syncthreads();

    const float a0 = k0[i], a1 = k1[i];
    int r0 = 0, r1 = 0;
    #pragma unroll 8
    for (int j = 0; j < Bn; ++j) {
        const float b0 = k0[j], b1 = k1[j];
        r0 += (b0 < a0) || (b0 == a0 && j < i);
        r1 += (b1 < a1) || (b1 == a1 && j < i);
    }
    inv1[r1] = i;
    __syncthreads();

    src[t * Bn + i] = inv1[r0];
    if (t == 0) y_out[i] = yb;   // pass labels through (as output dtype float)
}

// ---------------- gather/copy kernel (async-to-LDS pipeline) ----------------
__global__ void __launch_bounds__(256)
temporalmix_gather_kernel(const v4f* __restrict__ xin,
                          const int* __restrict__ src,
                          v4f* __restrict__ xout) {
    // per-wave private staging ring: 8 waves * DEPTH slots * 32 lanes * 16B = 32KB
    __shared__ v4f stage[8 * DEPTH * 32];

    const int lane   = threadIdx.x & 31;
    const int wave   = threadIdx.x >> 5;
    const int gwave  = blockIdx.x * (blockDim.x >> 5) + wave;
    const int nwaves = gridDim.x * (blockDim.x >> 5);

    v4f* wstage = &stage[wave * DEPTH * 32];
    // 32-bit LDS byte offset of this wave's staging base
    const uint32_t ldsBase =
        (uint32_t)(uintptr_t)(__attribute__((address_space(3))) char*)(char*)wstage;

    auto issue = [&](int slot, int chunk) {
        const int row = chunk / CHUNKS_PER_ROW;
        const int k32 = chunk - row * CHUNKS_PER_ROW;
        const int b   = row >> 4;          // NROWS laid out as b*Tn + t
        const int t   = row & 15;
        const int s   = __builtin_amdgcn_readfirstlane(src[t * Bn + b]);
        const v4f* gp = xin + ((s * Tn + t) * ROW_V4 + k32 * 32 + lane);
        const uint32_t lp = ldsBase + (uint32_t)((slot * 32 + lane) * 16);
        // read-once stream: non-temporal load, bypass far-cache retention
        asm volatile("global_load_async_to_lds_b128 %0, %1, off th:TH_LOAD_NT"
                     :: "v"(lp), "v"(gp)
                     : "memory");
    };

    int it = gwave;   // next chunk to issue
    int nq = 0;
    for (; nq < DEPTH && it < NCHUNKS; ++nq, it += nwaves) issue(nq, it);

    int cons = gwave; // next chunk to consume (in issue order)
    int slot = 0;

    // steady state: exactly DEPTH loads in flight (ASYNCcnt tracks loads only;
    // output stores ride STOREcnt, so oldest-load-done == asynccnt <= DEPTH-1)
    while (it < NCHUNKS) {
        asm volatile("s_wait_asynccnt 7" ::: "memory");  // oldest load landed
        v4f v = wstage[slot * 32 + lane];                // ds_load_b128
        __builtin_nontemporal_store(v, &xout[cons * 32 + lane]); // NT b128 store
        issue(slot, it);                                 // refill freed slot
        it   += nwaves;
        cons += nwaves;
        slot  = (slot + 1) & (DEPTH - 1);
    }

    // drain remaining in-flight loads
    asm volatile("s_wait_asynccnt 0" ::: "memory");
    for (; cons < NCHUNKS; cons += nwaves) {
        v4f v = wstage[slot * 32 + lane];
        __builtin_nontemporal_store(v, &xout[cons * 32 + lane]);
        slot = (slot + 1) & (DEPTH - 1);
    }
}

extern "C" void kernel_launch(void* const* d_in, const int* in_sizes, int n_in,
                              void* d_out, int out_size, void* d_ws, size_t ws_size,
                              hipStream_t stream) {
    const float* x = (const float*)d_in[0];
    const int*   y = (const int*)d_in[1];

    float* out   = (float*)d_out;
    float* y_out = out + (size_t)NROWS * CHW;  // y follows x_mixed in flat output
    int*   src   = (int*)d_ws;                 // [Tn * Bn] ints of scratch

    temporalmix_perm_kernel<<<Tn, Bn, 0, stream>>>(y, src, y_out);

    temporalmix_gather_kernel<<<1024, 256, 0, stream>>>(
        (const v4f*)x, src, (v4f*)out);
}